// BlobDiceLoss_2456721293567
// MI455X (gfx1250) — compile-verified
//
#include <hip/hip_runtime.h>
#include <hip/hip_bf16.h>

// Problem constants (from reference)
constexpr int Bq   = 2;
constexpr int Cq   = 4;
constexpr int Dq   = 160;
constexpr int NB1q = 65;
constexpr int NBINS = Bq * Cq * NB1q;      // 520 (bin = (b*C + c)*NB1 + blob)
constexpr float SMOOTHq = 1e-6f;

// ---------------------------------------------------------------------------
// Kernel 0: zero the global accumulator workspace (sums + counts)
// ---------------------------------------------------------------------------
__global__ void blob_zero_ws_kernel(float* __restrict__ ws, int n) {
    int i = blockIdx.x * blockDim.x + threadIdx.x;
    if (i < n) ws[i] = 0.0f;
}

// ---------------------------------------------------------------------------
// Kernel 1: streaming segmented reduction.
// One thread owns one contiguous W-row of 160 elements (float4/int4 loads).
// Labels are contiguous blob runs -> run-length compress before touching LDS.
// Per-block 520-bin LDS histogram, flushed via global float atomics.
// ---------------------------------------------------------------------------
__global__ __launch_bounds__(256) void blob_accum_kernel(
    const float* __restrict__ x, const int* __restrict__ labels,
    float* __restrict__ gsum, float* __restrict__ gcnt) {

    __shared__ float lsum[NBINS];
    __shared__ float lcnt[NBINS];
    for (int i = threadIdx.x; i < NBINS; i += 256) { lsum[i] = 0.0f; lcnt[i] = 0.0f; }
    __syncthreads();

    const int  row     = blockIdx.x * 256 + threadIdx.x;      // 0 .. 204799
    const int  bc      = row / (Dq * Dq);                     // batch*C + class
    const int  binBase = bc * NB1q;
    const long base    = (long)row * Dq;                      // 640B aligned
    const float4* __restrict__ xr = (const float4*)(x + base);
    const int4*   __restrict__ lr = (const int4*)(labels + base);

    int   curLab = -1;
    int   curCnt = 0;
    float curSum = 0.0f;

    #pragma unroll 4
    for (int i = 0; i < Dq / 4; ++i) {
        float4 xv = xr[i];
        int4   lv = lr[i];
        if (i + 8 < Dq / 4) {                 // ~128B ahead -> global_prefetch_b8
            __builtin_prefetch((const void*)&xr[i + 8], 0, 0);
            __builtin_prefetch((const void*)&lr[i + 8], 0, 0);
        }
        const int   la[4] = {lv.x, lv.y, lv.z, lv.w};
        const float va[4] = {xv.x, xv.y, xv.z, xv.w};
        #pragma unroll
        for (int j = 0; j < 4; ++j) {
            if (la[j] == curLab) {
                curSum += va[j]; ++curCnt;
            } else {
                if (curCnt) {
                    atomicAdd(&lsum[binBase + curLab], curSum);
                    atomicAdd(&lcnt[binBase + curLab], (float)curCnt);
                }
                curLab = la[j]; curSum = va[j]; curCnt = 1;
            }
        }
    }
    if (curCnt) {
        atomicAdd(&lsum[binBase + curLab], curSum);
        atomicAdd(&lcnt[binBase + curLab], (float)curCnt);
    }
    __syncthreads();

    for (int i = threadIdx.x; i < NBINS; i += 256) {
        if (lcnt[i] != 0.0f) {
            atomicAdd(&gsum[i], lsum[i]);
            atomicAdd(&gcnt[i], lcnt[i]);
        }
    }
}

// ---------------------------------------------------------------------------
// Kernel 2: dice epilogue on 520 bins, single wave32.
// Cross-lane sums use V_WMMA_F32_16X16X4_F32 with an all-ones B matrix:
// D[m,n] = sum_k A[m,k]  (exact f32, RNE). One WMMA + one shfl_xor(16)
// reduces all 32 lanes.
// ---------------------------------------------------------------------------
typedef __attribute__((ext_vector_type(2))) float v2f;
typedef __attribute__((ext_vector_type(8))) float v8f;

__device__ inline float wave_sum_wmma(float v) {
    v2f a; a.x = v;    a.y = 0.0f;    // A(16x4): row m = [v(m),0,v(m+16),0]
    v2f b; b.x = 1.0f; b.y = 1.0f;    // B(4x16) = all ones (layout-agnostic)
    v8f c = {};
    v8f d = __builtin_amdgcn_wmma_f32_16x16x4_f32(
        /*neg_a=*/false, a, /*neg_b=*/false, b,
        /*c_mod=*/(short)0, c, /*reuse_a=*/false, /*reuse_b=*/false);
    // lanes 0-15 hold M=0..7 partials, lanes 16-31 hold M=8..15 partials
    float s = d[0] + d[1] + d[2] + d[3] + d[4] + d[5] + d[6] + d[7];
    s += __shfl_xor(s, 16, 32);
    return s;                          // uniform across the wave
}

__global__ void blob_finalize_kernel(const float* __restrict__ gsum,
                                     const float* __restrict__ gcnt,
                                     float* __restrict__ out) {
    const int lane = threadIdx.x;      // launched with exactly 32 threads
    float accD[Bq] = {0.0f, 0.0f};
    float accN[Bq] = {0.0f, 0.0f};
    for (int idx = lane; idx < NBINS; idx += 32) {
        const int b  = idx / (Cq * NB1q);
        const int c  = (idx / NB1q) % Cq;
        const int nb = idx % NB1q;
        const float s = gsum[idx];
        const float n = gcnt[idx];
        if (n > 0.0f && nb >= 1 && c >= 1) {
            const float dice = (2.0f * s + SMOOTHq) / (s + n + SMOOTHq);
            accD[b] += dice;
            accN[b] += 1.0f;
        }
    }
    // Reconverged here: EXEC all ones for WMMA.
    const float d0 = wave_sum_wmma(accD[0]);
    const float n0 = wave_sum_wmma(accN[0]);
    const float d1 = wave_sum_wmma(accD[1]);
    const float n1 = wave_sum_wmma(accN[1]);
    if (lane == 0) {
        const float l0 = (n0 > 0.0f) ? -(d0 / fmaxf(n0, 1.0f)) : 0.0f;
        const float l1 = (n1 > 0.0f) ? -(d1 / fmaxf(n1, 1.0f)) : 0.0f;
        out[0] = 0.5f * (l0 + l1);
    }
}

// ---------------------------------------------------------------------------
extern "C" void kernel_launch(void* const* d_in, const int* in_sizes, int n_in,
                              void* d_out, int out_size, void* d_ws, size_t ws_size,
                              hipStream_t stream) {
    const float* x      = (const float*)d_in[0];   // [B,C,D,D,D] f32
    // d_in[1] = y (int64) — unused by the reference computation
    const int*   labels = (const int*)d_in[2];     // [B,C,D,D,D] i32

    float* gsum = (float*)d_ws;        // [520]
    float* gcnt = gsum + NBINS;        // [520]

    blob_zero_ws_kernel<<<(2 * NBINS + 255) / 256, 256, 0, stream>>>(gsum, 2 * NBINS);

    const int nrows = Bq * Cq * Dq * Dq;           // 204800 -> 800 blocks
    blob_accum_kernel<<<nrows / 256, 256, 0, stream>>>(x, labels, gsum, gcnt);

    blob_finalize_kernel<<<1, 32, 0, stream>>>(gsum, gcnt, (float*)d_out);
}